// ArtemisNet_64295660421275
// MI455X (gfx1250) — compile-verified
//
#include <hip/hip_runtime.h>
#include <hip/hip_bf16.h>

typedef __attribute__((ext_vector_type(2))) float v2f;
typedef __attribute__((ext_vector_type(8))) float v8f;

#define EPSBN 1e-5f

// ---------------- utility kernels ----------------

__global__ void zero_kernel(float* __restrict__ p, size_t n) {
  size_t i = (size_t)blockIdx.x * blockDim.x + threadIdx.x;
  if (i < n) p[i] = 0.0f;
}

__global__ void count_kernel(const int* __restrict__ ei, float* __restrict__ cnt, int E_) {
  int e = blockIdx.x * blockDim.x + threadIdx.x;
  if (e < E_) atomicAdd(&cnt[ei[E_ + e]], 1.0f);
}

// scatter-add h[src] into sum[dst]; 32 threads per edge, float4 per thread
__global__ void scatter_kernel(const float* __restrict__ h, const int* __restrict__ ei,
                               float* __restrict__ sum, int E_) {
  int tid = blockIdx.x * blockDim.x + threadIdx.x;
  if (tid >= E_ * 32) return;
  int e = tid >> 5;
  int c = tid & 31;
  int s = ei[e];
  int d = ei[E_ + e];
  const float4 v = *(const float4*)(h + (size_t)s * 128 + c * 4);
  float* o = sum + (size_t)d * 128 + c * 4;
  atomicAdd(o + 0, v.x);
  atomicAdd(o + 1, v.y);
  atomicAdd(o + 2, v.z);
  atomicAdd(o + 3, v.w);
}

// ---------------- conv1: per-edge message GEMM via fp32 WMMA ----------------
// msg[e,:] = [x[src[e]] (64) , ea[e] (32)] @ W1^T ; scatter-add into sum[dst]
__global__ __launch_bounds__(256) void conv1_edge(
    const float* __restrict__ x, const float* __restrict__ ea,
    const int* __restrict__ ei, const float* __restrict__ W1,
    float* __restrict__ sum, int E_) {
  const int wave = (blockIdx.x * blockDim.x + threadIdx.x) >> 5;
  const int ebase = wave * 16;
  if (ebase >= E_) return;                 // wave-uniform exit (EXEC stays all-ones)
  const int lane = threadIdx.x & 31;
  const int row = lane & 15;
  const int hk = lane >> 4;                 // 0: K={0,1}, 1: K={2,3} within a chunk

  int erow = ebase + row;
  if (erow >= E_) erow = E_ - 1;            // clamp; bad rows masked at scatter
  const int s = ei[erow];
  const float* xrow = x + (size_t)s * 64;
  const float* earow = ea + (size_t)erow * 32;

  v8f acc[8];
#pragma unroll
  for (int t = 0; t < 8; ++t) acc[t] = (v8f){0.f, 0.f, 0.f, 0.f, 0.f, 0.f, 0.f, 0.f};

  // K = 0..63 : x[src] part
  for (int c = 0; c < 16; ++c) {
    const int kb = c * 4 + hk * 2;
    v2f a = *(const v2f*)(xrow + kb);
#pragma unroll
    for (int t = 0; t < 8; ++t) {
      v2f b = *(const v2f*)(W1 + (t * 16 + row) * 96 + kb);
      acc[t] = __builtin_amdgcn_wmma_f32_16x16x4_f32(false, a, false, b,
                                                     (short)0, acc[t], false, false);
    }
  }
  // K = 64..95 : edge_attr part
  for (int c = 0; c < 8; ++c) {
    const int kb = c * 4 + hk * 2;
    v2f a = *(const v2f*)(earow + kb);
#pragma unroll
    for (int t = 0; t < 8; ++t) {
      v2f b = *(const v2f*)(W1 + (t * 16 + row) * 96 + 64 + kb);
      acc[t] = __builtin_amdgcn_wmma_f32_16x16x4_f32(false, a, false, b,
                                                     (short)0, acc[t], false, false);
    }
  }

  // scatter: acc[t][r] is (edge m = r + hk*8, col n = t*16 + row)
  const int* dstp = ei + E_;
  int dsts[8];
#pragma unroll
  for (int r = 0; r < 8; ++r) {
    int e = ebase + r + hk * 8;
    dsts[r] = (e < E_) ? dstp[e] : -1;
  }
#pragma unroll
  for (int t = 0; t < 8; ++t) {
#pragma unroll
    for (int r = 0; r < 8; ++r) {
      if (dsts[r] >= 0)
        atomicAdd(sum + (size_t)dsts[r] * 128 + t * 16 + row, acc[t][r]);
    }
  }
}

// conv1 node epilogue: mean, (+b1 only when deg>0), relu, BN, relu
__global__ void conv1_node(const float* __restrict__ sum, const float* __restrict__ cnt,
                           const float* __restrict__ b1,
                           const float* __restrict__ g, const float* __restrict__ be,
                           const float* __restrict__ mn, const float* __restrict__ vr,
                           float* __restrict__ hout, int total) {
  int tid = blockIdx.x * blockDim.x + threadIdx.x;
  if (tid >= total) return;
  int n = tid >> 7, d = tid & 127;
  float c = cnt[n];
  float h = (c > 0.0f) ? (sum[tid] / c + b1[d]) : 0.0f;
  h = fmaxf(h, 0.0f);
  h = g[d] * (h - mn[d]) * rsqrtf(vr[d] + EPSBN) + be[d];
  hout[tid] = fmaxf(h, 0.0f);
}

// ---------------- SAGE node GEMMs via fp32 WMMA ----------------
// hout = relu(BN( (sum/max(cnt,1)) @ Wl^T + bl + hprev @ Wr^T ))
__global__ __launch_bounds__(256) void sage_node(
    const float* __restrict__ sum, const float* __restrict__ cnt,
    const float* __restrict__ hprev,
    const float* __restrict__ Wl, const float* __restrict__ bl,
    const float* __restrict__ Wr,
    const float* __restrict__ g, const float* __restrict__ be,
    const float* __restrict__ mn, const float* __restrict__ vr,
    float* __restrict__ hout, int N_) {
  const int wave = (blockIdx.x * blockDim.x + threadIdx.x) >> 5;
  const int nbase = wave * 16;
  if (nbase >= N_) return;
  const int lane = threadIdx.x & 31;
  const int row = lane & 15;
  const int hk = lane >> 4;

  int nrow = nbase + row;
  if (nrow >= N_) nrow = N_ - 1;
  const float inv = 1.0f / fmaxf(cnt[nrow], 1.0f);
  const float* aggrow = sum + (size_t)nrow * 128;
  const float* selfrow = hprev + (size_t)nrow * 128;

  v8f acc[8];
#pragma unroll
  for (int t = 0; t < 8; ++t) acc[t] = (v8f){0.f, 0.f, 0.f, 0.f, 0.f, 0.f, 0.f, 0.f};

  // pass 1: agg @ Wl^T
  for (int c = 0; c < 32; ++c) {
    const int kb = c * 4 + hk * 2;
    v2f a = *(const v2f*)(aggrow + kb) * inv;
#pragma unroll
    for (int t = 0; t < 8; ++t) {
      v2f b = *(const v2f*)(Wl + (t * 16 + row) * 128 + kb);
      acc[t] = __builtin_amdgcn_wmma_f32_16x16x4_f32(false, a, false, b,
                                                     (short)0, acc[t], false, false);
    }
  }
  // pass 2: self @ Wr^T
  for (int c = 0; c < 32; ++c) {
    const int kb = c * 4 + hk * 2;
    v2f a = *(const v2f*)(selfrow + kb);
#pragma unroll
    for (int t = 0; t < 8; ++t) {
      v2f b = *(const v2f*)(Wr + (t * 16 + row) * 128 + kb);
      acc[t] = __builtin_amdgcn_wmma_f32_16x16x4_f32(false, a, false, b,
                                                     (short)0, acc[t], false, false);
    }
  }

  // epilogue: bias + BN + relu, store
#pragma unroll
  for (int t = 0; t < 8; ++t) {
    const int ncol = t * 16 + row;
    const float blv = bl[ncol];
    const float gv = g[ncol];
    const float bev = be[ncol];
    const float mv = mn[ncol];
    const float rv = rsqrtf(vr[ncol] + EPSBN);
#pragma unroll
    for (int r = 0; r < 8; ++r) {
      int node = nbase + r + hk * 8;
      if (node < N_) {
        float val = acc[t][r] + blv;
        val = gv * (val - mv) * rv + bev;
        hout[(size_t)node * 128 + ncol] = fmaxf(val, 0.0f);
      }
    }
  }
}

// ---------------- readout MLP (scalar; weights are wave-uniform -> scalar cached) ----------
__global__ __launch_bounds__(256) void mlp_kernel(
    const float* __restrict__ h, const float* __restrict__ x,
    const float* __restrict__ W4, const float* __restrict__ b4,
    const float* __restrict__ W5, const float* __restrict__ b5,
    float* __restrict__ out, int N_) {
  int n = blockIdx.x * blockDim.x + threadIdx.x;
  if (n >= N_) return;
  float z[64];
#pragma unroll
  for (int j = 0; j < 64; ++j) z[j] = b4[j];
  const float* hr = h + (size_t)n * 128;
  const float* xr = x + (size_t)n * 64;
  for (int k = 0; k < 128; ++k) {
    float in = hr[k];
#pragma unroll
    for (int j = 0; j < 64; ++j) z[j] = fmaf(in, W4[j * 192 + k], z[j]);
  }
  for (int k = 0; k < 64; ++k) {
    float in = xr[k];
#pragma unroll
    for (int j = 0; j < 64; ++j) z[j] = fmaf(in, W4[j * 192 + 128 + k], z[j]);
  }
  float o = b5[0];
#pragma unroll
  for (int j = 0; j < 64; ++j) o += fmaxf(z[j], 0.0f) * W5[j];
  out[n] = o;
}

// ---------------- host-side launcher ----------------

extern "C" void kernel_launch(void* const* d_in, const int* in_sizes, int n_in,
                              void* d_out, int out_size, void* d_ws, size_t ws_size,
                              hipStream_t stream) {
  (void)n_in; (void)out_size; (void)ws_size;

  const float* x   = (const float*)d_in[0];
  const int*   ei0 = (const int*)d_in[1];
  const int*   ei1 = (const int*)d_in[2];
  const int*   ei2 = (const int*)d_in[3];
  const float* ea0 = (const float*)d_in[4];
  const float* W1  = (const float*)d_in[5];
  const float* b1  = (const float*)d_in[6];
  const float* g1  = (const float*)d_in[7];
  const float* be1 = (const float*)d_in[8];
  const float* m1  = (const float*)d_in[9];
  const float* v1  = (const float*)d_in[10];
  const float* Wl2 = (const float*)d_in[11];
  const float* bl2 = (const float*)d_in[12];
  const float* Wr2 = (const float*)d_in[13];
  const float* g2  = (const float*)d_in[14];
  const float* be2 = (const float*)d_in[15];
  const float* m2  = (const float*)d_in[16];
  const float* v2  = (const float*)d_in[17];
  const float* Wl3 = (const float*)d_in[18];
  const float* bl3 = (const float*)d_in[19];
  const float* Wr3 = (const float*)d_in[20];
  const float* g3  = (const float*)d_in[21];
  const float* be3 = (const float*)d_in[22];
  const float* m3  = (const float*)d_in[23];
  const float* v3  = (const float*)d_in[24];
  const float* W4  = (const float*)d_in[25];
  const float* b4  = (const float*)d_in[26];
  const float* W5  = (const float*)d_in[27];
  const float* b5  = (const float*)d_in[28];

  const int N_ = in_sizes[0] / 64;   // 50000
  const int E_ = in_sizes[1] / 2;    // 800000

  // workspace layout (floats): cnt0 | cnt1 | cnt2 | sum[N*128] | hA[N*128] | hB[N*128]
  float* cnt0 = (float*)d_ws;
  float* cnt1 = cnt0 + N_;
  float* cnt2 = cnt1 + N_;
  float* sum  = cnt2 + N_;
  float* hA   = sum + (size_t)N_ * 128;
  float* hB   = hA + (size_t)N_ * 128;

  const int B = 256;
  const size_t nsum = (size_t)N_ * 128;

  // zero cnt0..2 + sum in one shot (contiguous)
  {
    size_t nz = (size_t)N_ * 3 + nsum;
    zero_kernel<<<(unsigned)((nz + B - 1) / B), B, 0, stream>>>(cnt0, nz);
  }

  // degrees
  count_kernel<<<(E_ + B - 1) / B, B, 0, stream>>>(ei0, cnt0, E_);
  count_kernel<<<(E_ + B - 1) / B, B, 0, stream>>>(ei1, cnt1, E_);
  count_kernel<<<(E_ + B - 1) / B, B, 0, stream>>>(ei2, cnt2, E_);

  // conv1
  {
    int waves = (E_ + 15) / 16;
    conv1_edge<<<(waves * 32 + B - 1) / B, B, 0, stream>>>(x, ea0, ei0, W1, sum, E_);
    int total = N_ * 128;
    conv1_node<<<(total + B - 1) / B, B, 0, stream>>>(sum, cnt0, b1, g1, be1, m1, v1, hA, total);
  }

  // conv2
  zero_kernel<<<(unsigned)((nsum + B - 1) / B), B, 0, stream>>>(sum, nsum);
  scatter_kernel<<<(E_ * 32 + B - 1) / B, B, 0, stream>>>(hA, ei1, sum, E_);
  {
    int waves = (N_ + 15) / 16;
    sage_node<<<(waves * 32 + B - 1) / B, B, 0, stream>>>(
        sum, cnt1, hA, Wl2, bl2, Wr2, g2, be2, m2, v2, hB, N_);
  }

  // conv3
  zero_kernel<<<(unsigned)((nsum + B - 1) / B), B, 0, stream>>>(sum, nsum);
  scatter_kernel<<<(E_ * 32 + B - 1) / B, B, 0, stream>>>(hB, ei2, sum, E_);
  {
    int waves = (N_ + 15) / 16;
    sage_node<<<(waves * 32 + B - 1) / B, B, 0, stream>>>(
        sum, cnt2, hB, Wl3, bl3, Wr3, g3, be3, m3, v3, hA, N_);
  }

  // readout
  mlp_kernel<<<(N_ + B - 1) / B, B, 0, stream>>>(hA, x, W4, b4, W5, b5, (float*)d_out, N_);
}